// TimeTempTransformerModule_41996190220747
// MI455X (gfx1250) — compile-verified
//
#include <hip/hip_runtime.h>

typedef __attribute__((ext_vector_type(2))) float v2f;
typedef __attribute__((ext_vector_type(8))) float v8f;

#define SEQL  32
#define DHID  32
#define BT    16        // batch tile per wave == WMMA N
#define WAVES 4
#define SSTR  34        // s-stride in feat LDS (even -> 8B aligned float2 reads)
#define BSTR  1092      // batch stride (32*34+4), == 4 mod 64 -> conflict-free frag reads
#define FEATW (BT*BSTR) // 17472 floats per wave
#define WOFF  576       // staged featurizer weights: w1(16)+b1(16)+b2(32)+w2(512)

static __device__ __forceinline__ float sigm(float x) {
  return __builtin_amdgcn_rcpf(1.f + __expf(-x));
}
static __device__ __forceinline__ float tanh_fast(float x) {
  // 1 - 2/(e^{2x}+1): monotone-safe at both infinities
  return 1.f - 2.f * __builtin_amdgcn_rcpf(1.f + __expf(2.f * x));
}

// Build per-chunk B fragments (K=dim, N=batch) from a vector held in WMMA
// D layout: H[j*8+v] = val[dim = 16j + 8*hi + v][batch = lo].
// Chunk k needs (val[4k+2hi], val[4k+2hi+1]) for batch lo; every cross-lane
// move is exactly a lane <-> lane^16 swap.
static __device__ __forceinline__ void build_frags(const float* H, float* f0, float* f1, int hi) {
#pragma unroll
  for (int k = 0; k < 8; ++k) {
    const int j = k >> 2;
    const int c = 4 * (k & 1);
    float a0 = H[j*8+c],   a1 = H[j*8+c+1];
    float b0 = H[j*8+c+2], b1 = H[j*8+c+3];
    float s0 = hi ? a0 : b0;                 // hi half sends its (c,c+1), lo half its (c+2,c+3)
    float s1 = hi ? a1 : b1;
    float r0 = __shfl_xor(s0, 16, 32);
    float r1 = __shfl_xor(s1, 16, 32);
    if ((k & 3) < 2) { f0[k] = hi ? r0 : a0; f1[k] = hi ? r1 : a1; }
    else             { f0[k] = hi ? b0 : r0; f1[k] = hi ? b1 : r1; }
  }
}

// Accumulator init = bias vector broadcast into D layout: lane gets bias[16m+8hi+v].
static __device__ __forceinline__ v8f load_bias_tile(const float* __restrict__ b, int m, int hi) {
  const float* p = b + 16*m + 8*hi;
  v2f q0 = *(const v2f*)(p + 0);
  v2f q1 = *(const v2f*)(p + 2);
  v2f q2 = *(const v2f*)(p + 4);
  v2f q3 = *(const v2f*)(p + 6);
  v8f acc;
  acc[0]=q0[0]; acc[1]=q0[1]; acc[2]=q1[0]; acc[3]=q1[1];
  acc[4]=q2[0]; acc[5]=q2[1]; acc[6]=q3[0]; acc[7]=q3[1];
  return acc;
}

// One GRU step for a 16-batch tile: gates^T = W[96x32] * [feat|h]^T via
// v_wmma_f32_16x16x4_f32 with weights as A (M=gate rows), vectors as B (N=batch).
static __device__ __forceinline__ void gru_step(
    float* H, float* HS, const float* featw, int s,
    const float* __restrict__ w_ih, const float* __restrict__ w_hh,
    const float* __restrict__ b_ih, const float* __restrict__ b_hh,
    int lo, int hi) {
  // feat B fragments from LDS (8B-aligned, conflict-free)
  float ff0[8], ff1[8];
#pragma unroll
  for (int k = 0; k < 8; ++k) {
    v2f t = *(const v2f*)(featw + lo*BSTR + s*SSTR + 4*k + 2*hi);
    ff0[k] = t[0]; ff1[k] = t[1];
  }
  // hidden-state B fragments via half-wave swap (no LDS round trip)
  float fh0[8], fh1[8];
  build_frags(H, fh0, fh1, hi);

  v8f GX[6], GH[6];
#pragma unroll
  for (int m = 0; m < 6; ++m) {
    const float* wrI = w_ih + (16*m + lo) * DHID + 2*hi;   // A frag: row=gate, K-chunk cols
    const float* wrH = w_hh + (16*m + lo) * DHID + 2*hi;
    v8f ax = load_bias_tile(b_ih, m, hi);
    v8f ah = load_bias_tile(b_hh, m, hi);
#pragma unroll
    for (int k = 0; k < 8; ++k) {
      v2f aI = *(const v2f*)(wrI + 4*k);
      v2f aH = *(const v2f*)(wrH + 4*k);
      v2f bF; bF[0] = ff0[k]; bF[1] = ff1[k];
      v2f bH; bH[0] = fh0[k]; bH[1] = fh1[k];
      ax = __builtin_amdgcn_wmma_f32_16x16x4_f32(false, aI, false, bF, (short)0, ax, false, false);
      ah = __builtin_amdgcn_wmma_f32_16x16x4_f32(false, aH, false, bH, (short)0, ah, false, false);
    }
    GX[m] = ax; GH[m] = ah;
  }
  // gate math in D layout: lane holds (dim = 16j+8hi+v, batch = lo).
  // tiles 0,1 = r; 2,3 = z; 4,5 = n — perfectly aligned with H tiles.
#pragma unroll
  for (int j = 0; j < 2; ++j) {
#pragma unroll
    for (int v = 0; v < 8; ++v) {
      float r  = sigm(GX[j][v] + GH[j][v]);
      float z  = sigm(GX[2+j][v] + GH[2+j][v]);
      float n  = tanh_fast(GX[4+j][v] + r * GH[4+j][v]);
      float hN = n + z * (H[j*8+v] - n);   // (1-z)*n + z*h
      H[j*8+v]  = hN;
      HS[j*8+v] += hN;                     // running sum for mean pooling
    }
  }
}

__global__ __launch_bounds__(128) void bigru_fused_kernel(
    const float* __restrict__ x,
    const float* __restrict__ w1,     const float* __restrict__ b1,
    const float* __restrict__ w2,     const float* __restrict__ b2,
    const float* __restrict__ w_ih_f, const float* __restrict__ w_hh_f,
    const float* __restrict__ b_ih_f, const float* __restrict__ b_hh_f,
    const float* __restrict__ w_ih_b, const float* __restrict__ w_hh_b,
    const float* __restrict__ b_ih_b, const float* __restrict__ b_hh_b,
    const float* __restrict__ w_dec,  const float* __restrict__ b_dec,
    float* __restrict__ out) {
  extern __shared__ float smem[];
  const int tid   = threadIdx.x;
  const int wave  = tid >> 5;
  const int l     = tid & 31;
  const int lo    = l & 15;
  const int hi    = l >> 4;
  const int gbase = (blockIdx.x * WAVES + wave) * BT;
  float* featw = smem + WOFF + wave * FEATW;

  // stage featurizer weights into LDS for uniform broadcast reads
  for (int t = tid; t < WOFF; t += blockDim.x) {
    float v;
    if (t < 16)      v = w1[t];
    else if (t < 32) v = b1[t - 16];
    else if (t < 64) v = b2[t - 32];
    else             v = w2[t - 64];
    smem[t] = v;
  }
  __syncthreads();

  // ---- Stage 1: featurizer. Iter i handles batch gbase+i; lane l handles s=l.
  const float* w1s = smem;       const float* b1s = smem + 16;
  const float* b2s = smem + 32;  const float* w2s = smem + 64;
#pragma unroll 1
  for (int i = 0; i < BT; ++i) {
    float xv = x[(gbase + i) * SEQL + l];          // fully coalesced
    float hb[16];
#pragma unroll
    for (int j = 0; j < 16; ++j) {
      float t = xv * w1s[j] + b1s[j];
      hb[j] = t > 0.f ? t : 0.01f * t;             // leaky_relu 0.01
    }
    float* fo = featw + i * BSTR + l * SSTR;
#pragma unroll
    for (int d = 0; d < DHID; ++d) {
      float a = b2s[d];
#pragma unroll
      for (int j = 0; j < 16; ++j) a += w2s[d*16 + j] * hb[j];
      fo[d] = a;
    }
  }
  __syncthreads();

  // ---- Stage 2: fused bidirectional GRU scan (WMMA fp32).
  float HF[16], HSF[16], HB[16], HSB[16];
#pragma unroll
  for (int t = 0; t < 16; ++t) { HF[t]=0.f; HSF[t]=0.f; HB[t]=0.f; HSB[t]=0.f; }

#pragma unroll 1
  for (int s = 0; s < SEQL; ++s) {
    gru_step(HF, HSF, featw, s,          w_ih_f, w_hh_f, b_ih_f, b_hh_f, lo, hi);
    gru_step(HB, HSB, featw, SEQL-1-s,   w_ih_b, w_hh_b, b_ih_b, b_hh_b, lo, hi);
  }

  // ---- Stage 3: decoder. pooled = [mean_s h_f, mean_s h_b]  (64 dims -> 128).
#pragma unroll
  for (int t = 0; t < 16; ++t) { HSF[t] *= (1.f/SEQL); HSB[t] *= (1.f/SEQL); }
  float df0[16], df1[16];
  build_frags(HSF, df0,     df1,     hi);   // K chunks 0..7  (dims 0..31)
  build_frags(HSB, df0 + 8, df1 + 8, hi);   // K chunks 8..15 (dims 32..63)

#pragma unroll
  for (int m = 0; m < 8; ++m) {
    v8f acc = load_bias_tile(b_dec, m, hi);
    const float* wr = w_dec + (16*m + lo) * 64 + 2*hi;
#pragma unroll
    for (int k = 0; k < 16; ++k) {
      v2f a = *(const v2f*)(wr + 4*k);
      v2f bb; bb[0] = df0[k]; bb[1] = df1[k];
      acc = __builtin_amdgcn_wmma_f32_16x16x4_f32(false, a, false, bb, (short)0, acc, false, false);
    }
    float* op = out + (size_t)(gbase + lo) * 128 + 16*m + 8*hi;
#pragma unroll
    for (int v = 0; v < 8; ++v) op[v] = acc[v];    // merges to b128 stores (32B aligned)
  }
}

extern "C" void kernel_launch(void* const* d_in, const int* in_sizes, int n_in,
                              void* d_out, int out_size, void* d_ws, size_t ws_size,
                              hipStream_t stream) {
  const float* x      = (const float*)d_in[0];
  const float* w1     = (const float*)d_in[1];
  const float* b1     = (const float*)d_in[2];
  const float* w2     = (const float*)d_in[3];
  const float* b2     = (const float*)d_in[4];
  const float* w_ih_f = (const float*)d_in[5];
  const float* w_hh_f = (const float*)d_in[6];
  const float* b_ih_f = (const float*)d_in[7];
  const float* b_hh_f = (const float*)d_in[8];
  const float* w_ih_b = (const float*)d_in[9];
  const float* w_hh_b = (const float*)d_in[10];
  const float* b_ih_b = (const float*)d_in[11];
  const float* b_hh_b = (const float*)d_in[12];
  const float* w_dec  = (const float*)d_in[13];
  const float* b_dec  = (const float*)d_in[14];

  const int nbatch = in_sizes[0] / SEQL;          // 32768
  const int blocks = nbatch / (WAVES * BT);       // 512
  const size_t smem = (size_t)(WOFF + WAVES * FEATW) * sizeof(float); // 281,856 B (<320KB/WG)

  hipLaunchKernelGGL(bigru_fused_kernel, dim3(blocks), dim3(128), smem, stream,
                     x, w1, b1, w2, b2,
                     w_ih_f, w_hh_f, b_ih_f, b_hh_f,
                     w_ih_b, w_hh_b, b_ih_b, b_hh_b,
                     w_dec, b_dec, (float*)d_out);
}